// LocalPseudoFeatLoss_41472204210802
// MI455X (gfx1250) — compile-verified
//
#include <hip/hip_runtime.h>
#include <math.h>

// ---------------------------------------------------------------------------
// LocalPseudoFeatLoss for MI455X (gfx1250, wave32, WMMA)
// B=2, C=19 classes, CH=64 feature channels, H=W=128, K=7 (pad 3), TOPK=8
// Neighbor dot products are computed as spatial Gram matrices with
// V_WMMA_F32_16X16X4_F32 (exact f32), chained over K.
// ---------------------------------------------------------------------------

typedef __attribute__((ext_vector_type(2))) float v2f;
typedef __attribute__((ext_vector_type(8))) float v8f;

#define BATCH 2
#define NCLS 19
#define FCH 64
#define HH 128
#define WW 128
#define HWSZ (HH * WW)          // 16384
#define KK 7
#define K2 49
#define PADR 3
#define SEGW 16
#define PROWS 7
#define PCOLS 22                // 16 + 2*3
#define NPOS (PROWS * PCOLS)    // 154
#define PCH 20                  // 19 softmax channels + 1 zero pad
#define NSEG (BATCH * HH * (WW / SEGW))  // 2048
#define EPS 1e-8f

// ------------------------------ kernel 1: inverse norms --------------------
__global__ __launch_bounds__(256) void inorm_kernel(const float* __restrict__ xs,
                                                    const float* __restrict__ xe,
                                                    float* __restrict__ inorm_s,
                                                    float* __restrict__ inorm_e) {
  int t = blockIdx.x * blockDim.x + threadIdx.x;     // 0 .. 2*B*HW-1
  int f = t >> 15;                                    // 0=src 1=ema
  int pix = t & (BATCH * HWSZ - 1);                   // 0..32767
  int b = pix >> 14;
  int hw = pix & (HWSZ - 1);
  const float* x = f ? xe : xs;
  const float* base = x + (size_t)(b * FCH) * HWSZ + hw;
  float ss = 0.f;
#pragma unroll
  for (int c = 0; c < FCH; ++c) {
    float v = base[c * HWSZ];
    ss += v * v;
  }
  float r = 1.0f / fmaxf(sqrtf(ss), EPS);
  (f ? inorm_e : inorm_s)[pix] = r;
}

// ------------------------------ kernel 2: softmax probs --------------------
__global__ __launch_bounds__(256) void softmax_kernel(const float* __restrict__ logits,
                                                      float* __restrict__ pbuf) {
  int t = blockIdx.x * blockDim.x + threadIdx.x;     // 0 .. B*HW-1
  int b = t >> 14;
  int hw = t & (HWSZ - 1);
  const float* lg = logits + (size_t)(b * NCLS) * HWSZ + hw;
  float v[NCLS];
  float mx = -1e30f;
#pragma unroll
  for (int c = 0; c < NCLS; ++c) {
    v[c] = lg[c * HWSZ];
    mx = fmaxf(mx, v[c]);
  }
  float s = 0.f;
#pragma unroll
  for (int c = 0; c < NCLS; ++c) {
    v[c] = __expf(v[c] - mx);
    s += v[c];
  }
  float inv = 1.0f / s;
#pragma unroll
  for (int c = 0; c < NCLS; ++c)
    pbuf[(size_t)(b * NCLS + c) * HWSZ + hw] = v[c] * inv;
}

// ------------------------ device helper: decompose segment -----------------
__device__ inline void seg_decode(int seg, int& b, int& h, int& w0) {
  b = seg >> 10;                 // 128 rows * 8 segs = 1024 per batch
  int rr = seg & 1023;
  h = rr >> 3;
  w0 = (rr & 7) << 4;
}

// ---------------- kernel 3: source pos/neg local cosine sims ---------------
__global__ __launch_bounds__(32) void src_kernel(const float* __restrict__ x,
                                                 const int* __restrict__ gt,
                                                 const float* __restrict__ inorm,
                                                 float* __restrict__ part) {
  __shared__ __align__(16) float patch[NPOS * FCH];   // 38.5 KB
  __shared__ float inP[NPOS];
  __shared__ int gtP[NPOS];
  __shared__ float simL[SEGW * K2];

  int lane = threadIdx.x;
  int seg = blockIdx.x;
  int b, h, w0;
  seg_decode(seg, b, h, w0);

  // ---- stage feature patch (position-major, channel-contiguous) ----
  for (int e = lane; e < PROWS * FCH * PCOLS; e += 32) {
    int col = e % PCOLS;
    int tmp = e / PCOLS;
    int c = tmp % FCH;
    int r = tmp / FCH;
    int hh = h - PADR + r, ww = w0 - PADR + col;
    float v = 0.f;
    if (hh >= 0 && hh < HH && ww >= 0 && ww < WW)
      v = x[(size_t)((b * FCH + c) << 14) + (hh << 7) + ww];
    patch[(r * PCOLS + col) * FCH + c] = v;
  }
  for (int i = lane; i < NPOS; i += 32) {
    int r = i / PCOLS, col = i % PCOLS;
    int hh = h - PADR + r, ww = w0 - PADR + col;
    bool in = (hh >= 0 && hh < HH && ww >= 0 && ww < WW);
    inP[i] = in ? inorm[(b << 14) + (hh << 7) + ww] : 0.f;
    gtP[i] = in ? gt[(b << 14) + (hh << 7) + ww] : 0;  // jnp.pad -> label 0
  }
  __syncthreads();

  // ---- A fragments: 16 center vectors, all 16 K-steps (K=64) ----
  int m = lane & 15;
  bool hi = lane >= 16;
  int cbase = (PADR * PCOLS + PADR + m) * FCH + (hi ? 2 : 0);
  v2f af[16];
#pragma unroll
  for (int ck = 0; ck < 16; ++ck)
    af[ck] = *(const v2f*)&patch[cbase + ck * 4];

  // ---- Gram tiles: 7 neighbor rows x 2 col tiles ----
  int n = lane & 15;
  for (int r = 0; r < PROWS; ++r) {
    for (int t = 0; t < 2; ++t) {
      int col = t * 6 + n;
      int pb = (r * PCOLS + col) * FCH;
      v8f acc = {0.f, 0.f, 0.f, 0.f, 0.f, 0.f, 0.f, 0.f};
#pragma unroll
      for (int ck = 0; ck < 16; ++ck) {
        float4 q = *(const float4*)&patch[pb + ck * 4];
        v2f bb;
        bb.x = hi ? q.y : q.x;   // K rows 0/1
        bb.y = hi ? q.w : q.z;   // K rows 2/3
        acc = __builtin_amdgcn_wmma_f32_16x16x4_f32(
            false, af[ck], false, bb, (short)0, acc, false, false);
      }
      float icN = inP[r * PCOLS + col];
#pragma unroll
      for (int v = 0; v < 8; ++v) {
        int mm = v + (hi ? 8 : 0);
        int dw = col - mm;
        bool use = (t == 0) ? (mm <= 9) : (mm >= 10);
        if (use && dw >= 0 && dw <= 6) {
          float val = acc[v] * inP[PADR * PCOLS + PADR + mm] * icN;
          simL[mm * K2 + r * KK + dw] = val;
        }
      }
    }
  }
  __syncthreads();

  // ---- per-pixel pos/neg masked sums ----
  float sp = 0.f, sn = 0.f, cp = 0.f, cn = 0.f;
  if (lane < SEGW) {
    int mm = lane;
    int gtc = gtP[PADR * PCOLS + PADR + mm];
    if (gtc != 255) {
      for (int j = 0; j < K2; ++j) {
        int r = j / KK, dw = j % KK;
        int gn = gtP[r * PCOLS + mm + dw];
        float s = simL[mm * K2 + j];
        if (gn == gtc) { sp += s; cp += 1.f; }
        else           { sn += s; cn += 1.f; }
      }
    }
  }
  for (int off = 16; off; off >>= 1) {
    sp += __shfl_down(sp, off);
    cp += __shfl_down(cp, off);
    sn += __shfl_down(sn, off);
    cn += __shfl_down(cn, off);
  }
  if (lane == 0) {
    float* o = part + seg * 4;
    o[0] = sp; o[1] = cp; o[2] = sn; o[3] = cn;
  }
}

// ------------- kernel 4: target sims + cross agreement + top-k -------------
__global__ __launch_bounds__(32) void ema_kernel(const float* __restrict__ x,
                                                 const float* __restrict__ pbuf,
                                                 const float* __restrict__ inorm,
                                                 const float* __restrict__ mixm,
                                                 float* __restrict__ part) {
  __shared__ __align__(16) float patch[NPOS * FCH];   // 38.5 KB
  __shared__ __align__(16) float pP[NPOS * PCH];      // 12.3 KB
  __shared__ float inP[NPOS];
  __shared__ float simL[SEGW * K2];
  __shared__ float crossL[SEGW * K2];

  int lane = threadIdx.x;
  int seg = blockIdx.x;
  int b, h, w0;
  seg_decode(seg, b, h, w0);

  for (int e = lane; e < PROWS * FCH * PCOLS; e += 32) {
    int col = e % PCOLS;
    int tmp = e / PCOLS;
    int c = tmp % FCH;
    int r = tmp / FCH;
    int hh = h - PADR + r, ww = w0 - PADR + col;
    float v = 0.f;
    if (hh >= 0 && hh < HH && ww >= 0 && ww < WW)
      v = x[(size_t)((b * FCH + c) << 14) + (hh << 7) + ww];
    patch[(r * PCOLS + col) * FCH + c] = v;
  }
  for (int e = lane; e < PROWS * PCH * PCOLS; e += 32) {
    int col = e % PCOLS;
    int tmp = e / PCOLS;
    int c = tmp % PCH;
    int r = tmp / PCH;
    int hh = h - PADR + r, ww = w0 - PADR + col;
    float v = 0.f;
    if (c < NCLS && hh >= 0 && hh < HH && ww >= 0 && ww < WW)
      v = pbuf[(size_t)((b * NCLS + c) << 14) + (hh << 7) + ww];
    pP[(r * PCOLS + col) * PCH + c] = v;
  }
  for (int i = lane; i < NPOS; i += 32) {
    int r = i / PCOLS, col = i % PCOLS;
    int hh = h - PADR + r, ww = w0 - PADR + col;
    bool in = (hh >= 0 && hh < HH && ww >= 0 && ww < WW);
    inP[i] = in ? inorm[(b << 14) + (hh << 7) + ww] : 0.f;
  }
  __syncthreads();

  int m = lane & 15;
  bool hi = lane >= 16;
  int cbase = (PADR * PCOLS + PADR + m) * FCH + (hi ? 2 : 0);
  int cbase2 = (PADR * PCOLS + PADR + m) * PCH + (hi ? 2 : 0);
  v2f af[16];
  v2f ap[5];
#pragma unroll
  for (int ck = 0; ck < 16; ++ck) af[ck] = *(const v2f*)&patch[cbase + ck * 4];
#pragma unroll
  for (int ck = 0; ck < 5; ++ck) ap[ck] = *(const v2f*)&pP[cbase2 + ck * 4];

  int n = lane & 15;
  for (int r = 0; r < PROWS; ++r) {
    for (int t = 0; t < 2; ++t) {
      int col = t * 6 + n;
      int pb = (r * PCOLS + col) * FCH;
      int pb2 = (r * PCOLS + col) * PCH;
      // feature Gram (K=64)
      v8f acc = {0.f, 0.f, 0.f, 0.f, 0.f, 0.f, 0.f, 0.f};
#pragma unroll
      for (int ck = 0; ck < 16; ++ck) {
        float4 q = *(const float4*)&patch[pb + ck * 4];
        v2f bb;
        bb.x = hi ? q.y : q.x;
        bb.y = hi ? q.w : q.z;
        acc = __builtin_amdgcn_wmma_f32_16x16x4_f32(
            false, af[ck], false, bb, (short)0, acc, false, false);
      }
      // prob Gram (K=20, channel 19 is zero pad)
      v8f acp = {0.f, 0.f, 0.f, 0.f, 0.f, 0.f, 0.f, 0.f};
#pragma unroll
      for (int ck = 0; ck < 5; ++ck) {
        float4 q = *(const float4*)&pP[pb2 + ck * 4];
        v2f bb;
        bb.x = hi ? q.y : q.x;
        bb.y = hi ? q.w : q.z;
        acp = __builtin_amdgcn_wmma_f32_16x16x4_f32(
            false, ap[ck], false, bb, (short)0, acp, false, false);
      }
      float icN = inP[r * PCOLS + col];
      bool valid = icN != 0.f;
#pragma unroll
      for (int v = 0; v < 8; ++v) {
        int mm = v + (hi ? 8 : 0);
        int dw = col - mm;
        bool use = (t == 0) ? (mm <= 9) : (mm >= 10);
        if (use && dw >= 0 && dw <= 6) {
          int idx = mm * K2 + r * KK + dw;
          simL[idx] = acc[v] * inP[PADR * PCOLS + PADR + mm] * icN;
          // OOB neighbor: softmax(0 logits) = uniform -> dot = 1/19
          crossL[idx] = valid ? acp[v] : (1.0f / 19.0f);
        }
      }
    }
  }
  __syncthreads();

  // ---- per-pixel top-(TOPK+1)/bottom-TOPK selection ----
  float pacc = 0.f, nacc = 0.f, cnt = 0.f;
  if (lane < SEGW) {
    int mm = lane;
    float mix = mixm[(b << 14) + (h << 7) + w0 + mm];
    if (1.0f - mix > 0.5f) {
      cnt = 1.f;
      const float* S = &simL[mm * K2];
      const float* C = &crossL[mm * K2];
      unsigned long long used = 0ull;
      for (int t = 0; t < 9; ++t) {        // TOPK+1 most similar
        float best = -1e30f;
        int bj = 0;
        for (int j = 0; j < K2; ++j) {
          if ((used >> j) & 1ull) continue;
          float s = S[j];
          if (s > best) { best = s; bj = j; }
        }
        used |= 1ull << bj;
        pacc += best * (-C[bj]);
      }
      used = 0ull;
      for (int t = 0; t < 8; ++t) {        // TOPK least similar
        float best = 1e30f;
        int bj = 0;
        for (int j = 0; j < K2; ++j) {
          if ((used >> j) & 1ull) continue;
          float s = S[j];
          if (s < best) { best = s; bj = j; }
        }
        used |= 1ull << bj;
        nacc += (1.0f - best) * (-(1.0f - C[bj]));
      }
    }
  }
  for (int off = 16; off; off >>= 1) {
    pacc += __shfl_down(pacc, off);
    nacc += __shfl_down(nacc, off);
    cnt += __shfl_down(cnt, off);
  }
  if (lane == 0) {
    float* o = part + seg * 4;
    o[0] = pacc; o[1] = nacc; o[2] = cnt; o[3] = 0.f;
  }
}

// ------------------------------ kernel 5: finalize -------------------------
__global__ void final_kernel(const float* __restrict__ ps,
                             const float* __restrict__ pe,
                             float* __restrict__ out) {
  if (blockIdx.x == 0 && threadIdx.x == 0) {
    float Ssp = 0.f, Csp = 0.f, Ssn = 0.f, Csn = 0.f;
    float Sp = 0.f, Sn = 0.f, Cnt = 0.f;
    for (int s = 0; s < NSEG; ++s) {
      Ssp += ps[s * 4 + 0]; Csp += ps[s * 4 + 1];
      Ssn += ps[s * 4 + 2]; Csn += ps[s * 4 + 3];
    }
    for (int s = 0; s < NSEG; ++s) {
      Sp += pe[s * 4 + 0]; Sn += pe[s * 4 + 1]; Cnt += pe[s * 4 + 2];
    }
    out[0] = -(Ssp / fmaxf(Csp, 1.0f));
    out[1] = Ssn / fmaxf(Csn, 1.0f);
    out[2] = Sp / fmaxf(9.0f * Cnt, 1.0f);
    out[3] = Sn / fmaxf(8.0f * Cnt, 1.0f);
  }
}

// ------------------------------ launch ------------------------------------
extern "C" void kernel_launch(void* const* d_in, const int* in_sizes, int n_in,
                              void* d_out, int out_size, void* d_ws, size_t ws_size,
                              hipStream_t stream) {
  const float* logits = (const float*)d_in[0];
  const int* gt       = (const int*)d_in[1];
  const float* x_ema  = (const float*)d_in[2];
  const float* x_src  = (const float*)d_in[3];
  const float* mix    = (const float*)d_in[4];
  float* out = (float*)d_out;

  float* ws = (float*)d_ws;
  float* inorm_s = ws;                                  // 32768
  float* inorm_e = inorm_s + BATCH * HWSZ;              // 32768
  float* pbuf    = inorm_e + BATCH * HWSZ;              // 2*19*16384 = 622592
  float* part_s  = pbuf + (size_t)BATCH * NCLS * HWSZ;  // 2048*4
  float* part_e  = part_s + NSEG * 4;                   // 2048*4
  // total ws: ~2.82 MB

  inorm_kernel<<<(2 * BATCH * HWSZ) / 256, 256, 0, stream>>>(x_src, x_ema, inorm_s, inorm_e);
  softmax_kernel<<<(BATCH * HWSZ) / 256, 256, 0, stream>>>(logits, pbuf);
  src_kernel<<<NSEG, 32, 0, stream>>>(x_src, gt, inorm_s, part_s);
  ema_kernel<<<NSEG, 32, 0, stream>>>(x_ema, pbuf, inorm_e, mix, part_e);
  final_kernel<<<1, 32, 0, stream>>>(part_s, part_e, out);
}